// Bottom_Right_Corner_66623532695961
// MI455X (gfx1250) — compile-verified
//
#include <hip/hip_runtime.h>
#include <math.h>
#include <stdint.h>

// Problem shape (fixed by the reference): x[C=512, H=256, W=256] f32.
// out[c,h,w] = 2 * max over rectangle [0..h]x[0..w] of x[c,:,:]
#define CCH 512
#define HH  256
#define WW  256
#define ROWS 16                       // rows per pipelined group (16 KB contiguous)
#define NITER (HH / ROWS)
#define NWAVE 8                       // 256 threads / wave32
#define GROUP_BYTES (ROWS * WW * 4)   // 16 KB
#define NINF_I 0xff800000             // bit pattern of -inf

// v = max(v, v shifted up by N within each 16-lane DPP row); shifted-in = -inf.
#define ROW_SHR_MAX(v, CTRL)                                                     \
    do {                                                                         \
        int _s = __builtin_amdgcn_update_dpp((int)NINF_I, __float_as_int(v),     \
                                             (CTRL), 0xf, 0xf, false);           \
        (v) = fmaxf((v), __int_as_float(_s));                                    \
    } while (0)

// Guarded variant for the replicated 8-lane partial scan.
#define ROW_SHR_MAX_G(v, CTRL, GUARD)                                            \
    do {                                                                         \
        int _s = __builtin_amdgcn_update_dpp((int)NINF_I, __float_as_int(v),     \
                                             (CTRL), 0xf, 0xf, false);           \
        float _f = __int_as_float(_s);                                           \
        if (GUARD) (v) = fmaxf((v), _f);                                         \
    } while (0)

__global__ __launch_bounds__(256)
void corner_pool2d_kernel(const float* __restrict__ x, float* __restrict__ out)
{
    // Double-buffered 16-row staging (2 * 16 KB) + per-row wave partials.
    __shared__ float buf[2][ROWS * WW];
    __shared__ float partial[ROWS][NWAVE];

    const int t    = threadIdx.x;     // column 0..255
    const int lane = t & 31;
    const int wid  = t >> 5;
    const int l8   = lane & 7;
    const bool hi16 = (lane >= 16);
    const int c    = blockIdx.x;

    const float* xc = x   + (size_t)c * HH * WW;
    float*       oc = out + (size_t)c * HH * WW;

    // Each thread DMAs four coalesced 16-byte chunks per 16 KB group
    // (one per contiguous 4 KB slab).
    const int chunk16 = t * 16;

    // ---- preload group 0 into buf[0] via async global->LDS DMA ----
    {
        unsigned lb = (unsigned)(uintptr_t)(&buf[0][0]) + (unsigned)chunk16;
        #pragma unroll
        for (int q = 0; q < GROUP_BYTES / 4096; ++q) {
            unsigned l = lb + (unsigned)(q * 4096);
            int      g = chunk16 + q * 4096;
            asm volatile("global_load_async_to_lds_b128 %0, %1, %2 th:TH_LOAD_NT"
                         :: "v"(l), "v"(g), "s"(xc) : "memory");
        }
        asm volatile("s_wait_asynccnt 0x0" ::: "memory");
    }
    __syncthreads();

    float prev = -INFINITY;           // running 2-D max for this column
    int cur = 0;

    for (int it = 0; it < NITER; ++it) {
        const int nxt = cur ^ 1;

        // Prefetch the next 16-row group (overlaps with the scan below).
        if (it + 1 < NITER) {
            unsigned lb = (unsigned)(uintptr_t)(&buf[nxt][0]) + (unsigned)chunk16;
            int      gb = (it + 1) * GROUP_BYTES + chunk16;
            #pragma unroll
            for (int q = 0; q < GROUP_BYTES / 4096; ++q) {
                unsigned l = lb + (unsigned)(q * 4096);
                int      g = gb + q * 4096;
                asm volatile("global_load_async_to_lds_b128 %0, %1, %2 th:TH_LOAD_NT"
                             :: "v"(l), "v"(g), "s"(xc) : "memory");
            }
        }

        // Load the 16 column values for this thread.
        float v[ROWS];
        #pragma unroll
        for (int r = 0; r < ROWS; ++r)
            v[r] = buf[cur][r * WW + t];

        // Wave32 inclusive max-scan: DPP row_shr 1/2/4/8 inside 16-lane rows,
        // then permlanex16 folds lane15's total into lanes 16..31. Pure VALU.
        #pragma unroll
        for (int r = 0; r < ROWS; ++r) {
            ROW_SHR_MAX(v[r], 0x111);
            ROW_SHR_MAX(v[r], 0x112);
            ROW_SHR_MAX(v[r], 0x114);
            ROW_SHR_MAX(v[r], 0x118);
            int px = __builtin_amdgcn_permlanex16(__float_as_int(v[r]),
                                                  __float_as_int(v[r]),
                                                  -1, -1, false, false);
            if (hi16) v[r] = fmaxf(v[r], __int_as_float(px));
        }

        // Publish per-wave row totals.
        if (lane == 31) {
            #pragma unroll
            for (int r = 0; r < ROWS; ++r) partial[r][wid] = v[r];
        }
        __syncthreads();

        // Cross-wave combine: replicated 8-lane DPP scan of the partials,
        // then a uniform v_readlane broadcast of the prefix at wave (wid-1).
        const int src = (wid > 0) ? (wid - 1) : 0;
        #pragma unroll
        for (int r = 0; r < ROWS; ++r) {
            float p = partial[r][l8];
            ROW_SHR_MAX_G(p, 0x111, l8 >= 1);
            ROW_SHR_MAX_G(p, 0x112, l8 >= 2);
            ROW_SHR_MAX_G(p, 0x114, l8 >= 4);
            float off = __int_as_float(
                __builtin_amdgcn_readlane(__float_as_int(p), src));
            if (wid > 0) v[r] = fmaxf(v[r], off);
        }

        // Chain down rows and stream the doubled result out (non-temporal).
        const int hbase = it * ROWS;
        #pragma unroll
        for (int r = 0; r < ROWS; ++r) {
            prev = fmaxf(prev, v[r]);
            __builtin_nontemporal_store(prev + prev, &oc[(hbase + r) * WW + t]);
        }

        // Next group must be resident, and partials must be consumed,
        // before anyone advances.
        asm volatile("s_wait_asynccnt 0x0" ::: "memory");
        __syncthreads();
        cur = nxt;
    }
}

extern "C" void kernel_launch(void* const* d_in, const int* in_sizes, int n_in,
                              void* d_out, int out_size, void* d_ws, size_t ws_size,
                              hipStream_t stream) {
    (void)in_sizes; (void)n_in; (void)d_ws; (void)ws_size; (void)out_size;
    const float* x  = (const float*)d_in[0];
    float* out      = (float*)d_out;
    corner_pool2d_kernel<<<dim3(CCH), dim3(256), 0, stream>>>(x, out);
}